// MultipleEmbedding_75007308857671
// MI455X (gfx1250) — compile-verified
//
#include <hip/hip_runtime.h>
#include <hip/hip_bf16.h>

typedef __attribute__((ext_vector_type(2))) float v2f;
typedef __attribute__((ext_vector_type(8))) float v8f;

#define NTOK 65536
#define DIM 512
#define INW 2048
#define GRP 3
#define PER 5000
#define EPSLN 1e-5f
#define ROWS 32            // rows (tokens) per block = two 16-row M tiles
#define MAXTILES 2048      // NTOK/ROWS upper bound of row tiles per group
#define DIMP 516           // padded LDS row stride (bank-conflict-free frag reads)

// workspace layout (as ints): [0..2]=cnt[g], [4..6]=off[g], [16..16+NTOK)=list
// then (as floats) at index WS_PART_OFF: partial tile losses [GRP][MAXTILES]
#define WS_LIST_OFF 16
#define WS_PART_OFF (16 + NTOK)

// ---------------------------------------------------------------- zero output
__global__ void zero_kernel(float* __restrict__ out, size_t n) {
    size_t i = (size_t)blockIdx.x * blockDim.x + threadIdx.x;
    size_t stride = (size_t)gridDim.x * blockDim.x;
    for (; i < n; i += stride) out[i] = 0.0f;
}

// ------------------------------------------------- deterministic compaction
__global__ __launch_bounds__(1024)
void compact_kernel(const int* __restrict__ x, int* __restrict__ wsI) {
    __shared__ int sCnt[1024 * 3];
    __shared__ int sBase[3];
    const int t = threadIdx.x;
    const int base = t * 64;
    int c0 = 0, c1 = 0, c2 = 0;
    for (int i = 0; i < 64; ++i) {
        int xv = x[base + i];
        if (xv >= 1 && xv < 5001)            c0++;
        else if (xv >= 5001 && xv < 10001)   c1++;
        else if (xv >= 10001 && xv < 15001)  c2++;
    }
    sCnt[t * 3 + 0] = c0; sCnt[t * 3 + 1] = c1; sCnt[t * 3 + 2] = c2;
    __syncthreads();
    if (t == 0) {
        int r0 = 0, r1 = 0, r2 = 0;
        for (int i = 0; i < 1024; ++i) {
            int a = sCnt[i * 3 + 0], b = sCnt[i * 3 + 1], c = sCnt[i * 3 + 2];
            sCnt[i * 3 + 0] = r0; sCnt[i * 3 + 1] = r1; sCnt[i * 3 + 2] = r2;
            r0 += a; r1 += b; r2 += c;
        }
        wsI[0] = r0; wsI[1] = r1; wsI[2] = r2;
        wsI[4] = 0;  wsI[5] = r0; wsI[6] = r0 + r1;
        sBase[0] = 0; sBase[1] = r0; sBase[2] = r0 + r1;
    }
    __syncthreads();
    int p0 = sBase[0] + sCnt[t * 3 + 0];
    int p1 = sBase[1] + sCnt[t * 3 + 1];
    int p2 = sBase[2] + sCnt[t * 3 + 2];
    int* list = wsI + WS_LIST_OFF;
    for (int i = 0; i < 64; ++i) {
        int tok = base + i;
        int xv = x[tok];
        if (xv >= 1 && xv < 5001)            list[p0++] = tok;
        else if (xv >= 5001 && xv < 10001)   list[p1++] = tok;
        else if (xv >= 10001 && xv < 15001)  list[p2++] = tok;
    }
}

// --------------------------------------------------------------- fused core
// Block = 256 threads = 8 waves; ROWS=32 compacted rows (two M tiles).
// GEMM1 (enc = adj @ W^T) -> LDS, LayerNorm scatter, GEMM2 (recon) + loss.
__global__ __launch_bounds__(256)
void fused_kernel(const int* __restrict__ x,
                  const float* __restrict__ tables,
                  const float* __restrict__ W,
                  const float* __restrict__ bEnc,
                  const float* __restrict__ bDec,
                  const float* __restrict__ gam,
                  const float* __restrict__ bet,
                  float* __restrict__ out,
                  const int* __restrict__ wsI,
                  float* __restrict__ wsPart) {
    const int g    = blockIdx.y;
    const int cnt  = wsI[g];
    const int tile = blockIdx.x;
    if (tile * ROWS >= cnt) return;
    const int goff = wsI[4 + g];
    const int* list = wsI + WS_LIST_OFF;

    __shared__ int   sTok[ROWS];
    __shared__ int   sRowOff[ROWS];     // element offset of gathered row in tables
    __shared__ float sEnc[ROWS][DIMP];
    __shared__ float sErr[8][ROWS];
    __shared__ float sNnz[8][ROWS];
    __shared__ float sRowE[ROWS];

    const int tid = threadIdx.x;
    const int lo  = (g == 0) ? 0 : (g == 1 ? 5000 : 10000);
    if (tid < ROWS) {
        int r = tile * ROWS + tid;
        if (r < cnt) {
            int tok = list[goff + r];
            int lid = x[tok] - (lo + 1);
            sTok[tid]    = tok;
            sRowOff[tid] = (g * PER + lid) * INW;
        } else {
            sTok[tid]    = -1;
            sRowOff[tid] = (g * PER) * INW;   // safe dummy row
        }
    }
    __syncthreads();

    const float* Wg = W + (size_t)g * DIM * INW;
    const int wave = tid >> 5, lane = tid & 31;
    const int laneLo = lane & 15, laneHi = lane >> 4;

    // ---------------- Phase A: enc = adj @ W^T  (f32 WMMA, K = 2048) -------
    // Two M tiles (rows 0-15 and 16-31) share the four B fragments.
    const v8f vz = {0.f,0.f,0.f,0.f,0.f,0.f,0.f,0.f};
    v8f am0 = vz, am1 = vz, am2 = vz, am3 = vz;   // M tile 0 accumulators
    v8f bm0 = vz, bm1 = vz, bm2 = vz, bm3 = vz;   // M tile 1 accumulators
    const float* aP0 = tables + sRowOff[laneLo]      + 2 * laneHi;
    const float* aP1 = tables + sRowOff[16 + laneLo] + 2 * laneHi;
    const float* bP0 = Wg + (size_t)(wave * 64 +  0 + laneLo) * INW + 2 * laneHi;
    const float* bP1 = Wg + (size_t)(wave * 64 + 16 + laneLo) * INW + 2 * laneHi;
    const float* bP2 = Wg + (size_t)(wave * 64 + 32 + laneLo) * INW + 2 * laneHi;
    const float* bP3 = Wg + (size_t)(wave * 64 + 48 + laneLo) * INW + 2 * laneHi;
#pragma unroll 2
    for (int k = 0; k < INW; k += 4) {
        v2f a0 = *(const v2f*)(aP0 + k);
        v2f a1 = *(const v2f*)(aP1 + k);
        v2f b0 = *(const v2f*)(bP0 + k);
        v2f b1 = *(const v2f*)(bP1 + k);
        v2f b2 = *(const v2f*)(bP2 + k);
        v2f b3 = *(const v2f*)(bP3 + k);
        am0 = __builtin_amdgcn_wmma_f32_16x16x4_f32(false, a0, false, b0, (short)0, am0, false, false);
        bm0 = __builtin_amdgcn_wmma_f32_16x16x4_f32(false, a1, false, b0, (short)0, bm0, false, false);
        am1 = __builtin_amdgcn_wmma_f32_16x16x4_f32(false, a0, false, b1, (short)0, am1, false, false);
        bm1 = __builtin_amdgcn_wmma_f32_16x16x4_f32(false, a1, false, b1, (short)0, bm1, false, false);
        am2 = __builtin_amdgcn_wmma_f32_16x16x4_f32(false, a0, false, b2, (short)0, am2, false, false);
        bm2 = __builtin_amdgcn_wmma_f32_16x16x4_f32(false, a1, false, b2, (short)0, bm2, false, false);
        am3 = __builtin_amdgcn_wmma_f32_16x16x4_f32(false, a0, false, b3, (short)0, am3, false, false);
        bm3 = __builtin_amdgcn_wmma_f32_16x16x4_f32(false, a1, false, b3, (short)0, bm3, false, false);
    }
    {
        const float* bEncG = bEnc + g * DIM;
        v8f accA[4] = {am0, am1, am2, am3};
        v8f accB[4] = {bm0, bm1, bm2, bm3};
#pragma unroll
        for (int t = 0; t < 4; ++t) {
            int dcol = wave * 64 + t * 16 + laneLo;
            float be = bEncG[dcol];
#pragma unroll
            for (int j = 0; j < 8; ++j) {
                sEnc[j + 8 * laneHi][dcol]      = accA[t][j] + be;
                sEnc[16 + j + 8 * laneHi][dcol] = accB[t][j] + be;
            }
        }
    }
    __syncthreads();

    // ---------------- Phase B: LayerNorm(enc) -> scatter to out ------------
    {
        const int r = tid >> 3, c0 = tid & 7;       // 8 threads per row, 32 rows
        float s = 0.f;
        for (int i = 0; i < 64; ++i) s += sEnc[r][c0 + 8 * i];
        for (int m = 1; m < 8; m <<= 1) s += __shfl_xor(s, m, 32);
        float mu = s * (1.0f / 512.0f);
        float v = 0.f;
        for (int i = 0; i < 64; ++i) { float d = sEnc[r][c0 + 8 * i] - mu; v += d * d; }
        for (int m = 1; m < 8; m <<= 1) v += __shfl_xor(v, m, 32);
        float rs = rsqrtf(v * (1.0f / 512.0f) + EPSLN);
        int tok = sTok[r];
        if (tok >= 0) {
            const float* gG = gam + g * DIM;
            const float* bG = bet + g * DIM;
            float* orow = out + (size_t)tok * DIM;
            for (int i = 0; i < 64; ++i) {
                int c = c0 + 8 * i;
                orow[c] = (sEnc[r][c] - mu) * rs * gG[c] + bG[c];
            }
        }
    }

    // ---------------- Phase C: recon = enc @ W + b_dec; masked error -------
    const float* bDecG = bDec + (size_t)g * INW;
    float errA[8], nnzA[8], errB[8], nnzB[8];
#pragma unroll
    for (int j = 0; j < 8; ++j) { errA[j] = 0.f; nnzA[j] = 0.f; errB[j] = 0.f; nnzB[j] = 0.f; }

    for (int tt = 0; tt < 16; ++tt) {                 // wave covers 16 of 128 IN-tiles
        const int j0 = (wave * 16 + tt) * 16;
        v8f accA = vz, accB = vz;
        const float* bcol = Wg + j0 + laneLo;         // B[d][j] = W natural, coalesced
#pragma unroll 4
        for (int kd = 0; kd < DIM; kd += 4) {
            v2f a0 = *(const v2f*)&sEnc[laneLo][kd + 2 * laneHi];
            v2f a1 = *(const v2f*)&sEnc[16 + laneLo][kd + 2 * laneHi];
            v2f bv;
            bv.x = bcol[(size_t)(kd + 2 * laneHi) * INW];
            bv.y = bcol[(size_t)(kd + 1 + 2 * laneHi) * INW];
            accA = __builtin_amdgcn_wmma_f32_16x16x4_f32(false, a0, false, bv, (short)0, accA, false, false);
            accB = __builtin_amdgcn_wmma_f32_16x16x4_f32(false, a1, false, bv, (short)0, accB, false, false);
        }
        const float bd = bDecG[j0 + laneLo];
#pragma unroll
        for (int j = 0; j < 8; ++j) {
            int M0 = j + 8 * laneHi;
            float adj0 = tables[sRowOff[M0] + j0 + laneLo];
            float d0 = adj0 - (accA[j] + bd);
            if (adj0 != 0.0f) { errA[j] += d0 * d0; nnzA[j] += 1.0f; }
            float adj1 = tables[sRowOff[16 + M0] + j0 + laneLo];
            float d1 = adj1 - (accB[j] + bd);
            if (adj1 != 0.0f) { errB[j] += d1 * d1; nnzB[j] += 1.0f; }
        }
    }
#pragma unroll
    for (int j = 0; j < 8; ++j) {
        float e0 = errA[j], n0 = nnzA[j], e1 = errB[j], n1 = nnzB[j];
        for (int m = 1; m < 16; m <<= 1) {
            e0 += __shfl_xor(e0, m, 32); n0 += __shfl_xor(n0, m, 32);
            e1 += __shfl_xor(e1, m, 32); n1 += __shfl_xor(n1, m, 32);
        }
        if (laneLo == 0) {
            sErr[wave][j + 8 * laneHi] = e0;      sNnz[wave][j + 8 * laneHi] = n0;
            sErr[wave][16 + j + 8 * laneHi] = e1; sNnz[wave][16 + j + 8 * laneHi] = n1;
        }
    }
    __syncthreads();
    if (tid < ROWS) {
        float e = 0.f, n = 0.f;
        for (int w = 0; w < 8; ++w) { e += sErr[w][tid]; n += sNnz[w][tid]; }   // fixed order
        sRowE[tid] = (sTok[tid] >= 0) ? e / fmaxf(n, 1.0f) : 0.0f;
    }
    __syncthreads();
    if (tid == 0) {
        float s = 0.f;
        for (int r = 0; r < ROWS; ++r) s += sRowE[r];                            // fixed order
        wsPart[(size_t)g * MAXTILES + tile] = s;
    }
}

// --------------------------------------------------------- final loss reduce
__global__ void loss_kernel(const int* __restrict__ wsI,
                            const float* __restrict__ wsPart,
                            float* __restrict__ out) {
    if (threadIdx.x != 0 || blockIdx.x != 0) return;
    float total = 0.f;
    for (int g = 0; g < GRP; ++g) {
        int c = wsI[g];
        int tiles = (c + ROWS - 1) / ROWS;
        float s = 0.f;
        for (int t = 0; t < tiles; ++t) s += wsPart[(size_t)g * MAXTILES + t];
        if (c > 0) total += s / (float)c;
    }
    out[(size_t)NTOK * DIM] = total;
}

// ------------------------------------------------------------------- launch
extern "C" void kernel_launch(void* const* d_in, const int* in_sizes, int n_in,
                              void* d_out, int out_size, void* d_ws, size_t ws_size,
                              hipStream_t stream) {
    const int*   x      = (const int*)d_in[0];
    const float* tables = (const float*)d_in[1];
    const float* W      = (const float*)d_in[2];
    const float* bEnc   = (const float*)d_in[3];
    const float* bDec   = (const float*)d_in[4];
    const float* gam    = (const float*)d_in[5];
    const float* bet    = (const float*)d_in[6];
    float* out   = (float*)d_out;
    int*   wsI   = (int*)d_ws;
    float* wsPart = (float*)d_ws + WS_PART_OFF;

    zero_kernel<<<2048, 256, 0, stream>>>(out, (size_t)out_size);
    compact_kernel<<<1, 1024, 0, stream>>>(x, wsI);
    dim3 grid(MAXTILES, GRP);
    fused_kernel<<<grid, 256, 0, stream>>>(x, tables, W, bEnc, bDec, gam, bet,
                                           out, wsI, wsPart);
    loss_kernel<<<1, 64, 0, stream>>>(wsI, wsPart, out);
}